// MNNVLAllReduce_4492535792388
// MI455X (gfx1250) — compile-verified
//
#include <hip/hip_runtime.h>
#include <cstdint>
#include <cstddef>

// Problem constants (match reference)
#define TP      8
#define NTOK    2048
#define HID     8192
#define BLOCK   512                    // 16 wave32s per workgroup
#define EPT     (HID / (BLOCK * 4))    // float4 chunks per thread = 4
#define EPS_RMS 1e-6f

typedef float v4f __attribute__((ext_vector_type(4)));

__launch_bounds__(BLOCK, 2)
__global__ void fused_allreduce_resid_rmsnorm(
    const float* __restrict__ in,        // [TP, NTOK, HID]
    const float* __restrict__ residual,  // [NTOK, HID]
    const float* __restrict__ w,         // [HID]
    float* __restrict__ out,             // [NTOK, HID]
    float* __restrict__ res_out)         // [NTOK, HID]
{
    __shared__ __align__(16) float s_res[HID];       // 32 KiB staged residual row
    __shared__ float s_partial[BLOCK / 32];
    __shared__ float s_scale;

    const int tid   = threadIdx.x;
    const int token = blockIdx.x;
    const size_t row = (size_t)token * (size_t)HID;

    // ---- Stage residual row into LDS via CDNA5 async global->LDS loads ----
    // Each lane copies 4x16B; it later reads back exactly its own bytes, so a
    // per-wave s_wait_asynccnt 0 is sufficient (no cross-lane dependency).
    #pragma unroll
    for (int c = 0; c < EPT; ++c) {
        const int col = c * (BLOCK * 4) + tid * 4;
        uint32_t laddr = (uint32_t)(size_t)(const void*)&s_res[col];
        uint64_t gaddr = (uint64_t)(const void*)(residual + row + col);
        asm volatile("global_load_async_to_lds_b128 %0, %1, off"
                     :: "v"(laddr), "v"(gaddr) : "memory");
    }

    // ---- All-reduce over the TP axis straight into registers (overlaps) ----
    // Input is 512 MiB read exactly once (> 192 MB L2): non-temporal loads so
    // the stream doesn't evict the L2-resident norm_weight / other blocks' lines.
    v4f acc[EPT];
    #pragma unroll
    for (int c = 0; c < EPT; ++c) acc[c] = (v4f)(0.f);

    #pragma unroll
    for (int r = 0; r < TP; ++r) {
        const float* __restrict__ src = in + ((size_t)r * NTOK) * (size_t)HID + row;
        #pragma unroll
        for (int c = 0; c < EPT; ++c) {
            const int col = c * (BLOCK * 4) + tid * 4;
            const v4f v = __builtin_nontemporal_load((const v4f*)(src + col));
            acc[c] += v;
        }
    }

    // ---- Consume async tile: residual add, stream residual_out, local ssq ----
    asm volatile("s_wait_asynccnt 0x0" ::: "memory");
    float ssq = 0.f;
    #pragma unroll
    for (int c = 0; c < EPT; ++c) {
        const int col = c * (BLOCK * 4) + tid * 4;
        const v4f rv = *(const v4f*)&s_res[col];
        acc[c] += rv;
        // residual_out: written once, never re-read by this grid -> NT store
        __builtin_nontemporal_store(acc[c], (v4f*)(res_out + row + col));
        ssq += acc[c].x * acc[c].x + acc[c].y * acc[c].y
             + acc[c].z * acc[c].z + acc[c].w * acc[c].w;
    }

    // Weight loads stay regular-temporal: 32 KiB reused by all 2048 blocks (L2-hot)
    v4f wv[EPT];
    #pragma unroll
    for (int c = 0; c < EPT; ++c) {
        const int col = c * (BLOCK * 4) + tid * 4;
        wv[c] = *(const v4f*)(w + col);
    }

    // ---- Block reduction of sum-of-squares (wave32 shuffle tree + LDS) ----
    #pragma unroll
    for (int off = 16; off >= 1; off >>= 1)
        ssq += __shfl_xor(ssq, off, 32);

    const int lane = tid & 31;
    const int wid  = tid >> 5;
    if (lane == 0) s_partial[wid] = ssq;
    __syncthreads();

    if (tid < (BLOCK / 32)) {
        float v = s_partial[tid];
        #pragma unroll
        for (int off = (BLOCK / 64); off >= 1; off >>= 1)
            v += __shfl_xor(v, off, 32);
        if (tid == 0)
            s_scale = rsqrtf(v * (1.0f / (float)HID) + EPS_RMS);
    }
    __syncthreads();
    const float scale = s_scale;

    // ---- Normalize and store output (write-once -> NT store) ----
    #pragma unroll
    for (int c = 0; c < EPT; ++c) {
        const int col = c * (BLOCK * 4) + tid * 4;
        v4f o = acc[c] * scale * wv[c];
        __builtin_nontemporal_store(o, (v4f*)(out + row + col));
    }
}

extern "C" void kernel_launch(void* const* d_in, const int* in_sizes, int n_in,
                              void* d_out, int out_size, void* d_ws, size_t ws_size,
                              hipStream_t stream) {
    (void)in_sizes; (void)n_in; (void)out_size; (void)d_ws; (void)ws_size;

    const float* in       = (const float*)d_in[0];  // [TP, NTOK, HID]
    const float* residual = (const float*)d_in[1];  // [NTOK, HID]
    const float* w        = (const float*)d_in[2];  // [HID]

    float* out     = (float*)d_out;                         // output  [NTOK, HID]
    float* res_out = (float*)d_out + (size_t)NTOK * HID;    // residual_out [NTOK, HID]

    fused_allreduce_resid_rmsnorm<<<NTOK, BLOCK, 0, stream>>>(
        in, residual, w, out, res_out);
}